// DyConvAtten_6047313952930
// MI455X (gfx1250) — compile-verified
//
#include <hip/hip_runtime.h>
#include <hip/hip_bf16.h>

// DyConvAtten for MI455X (gfx1250, wave32, WMMA).
// Shapes: B=64, N=256, H=1024, K=3, NK=768.
//
// Stage 1: weight[b,m,o] = f[b,m,:] . W_lin[o,:] + b_lin[o]   (GEMM, fp32 WMMA)
// Stage 2: out[b,m,h]   = sum_{i,t} weight[b,m,i*3+t] * k[b,i,h+t-1]  (3 shifted GEMMs)
//          then LayerNorm over h.  Fused in one kernel (h kept in registers).

#define Bn 64
#define Nn 256
#define Hn 1024
#define NK 768
#define LN_EPS 1e-5f

typedef __attribute__((ext_vector_type(2))) float v2f;
typedef __attribute__((ext_vector_type(8))) float v8f;

// ---------------------------------------------------------------------------
// Kernel 1: weight = f @ W_lin^T + b_lin
// grid (o_tiles=6, m_tiles=4, b=64), block 256 (8 waves).
// Workgroup tile: M=64, N=128, Kstep=16. Wave tile: 16x64 (4 wmma accum).
// ---------------------------------------------------------------------------
__global__ __launch_bounds__(256) void wgen_gemm(
    const float* __restrict__ f, const float* __restrict__ Wlin,
    const float* __restrict__ blin, float* __restrict__ wout) {
  __shared__ float As[64 * 20];    // A tile, padded stride
  __shared__ float Bs[16 * 136];   // B tile (transposed W_lin), padded stride

  const int tid = threadIdx.x;
  const int b = blockIdx.z;
  const int m0 = blockIdx.y * 64;
  const int o0 = blockIdx.x * 128;
  const int wave = tid >> 5, lane = tid & 31;
  const int lm = lane & 15, hi = lane >> 4;
  const int wm = (wave & 3) * 16;   // wave m-offset in [0,64)
  const int wn = (wave >> 2) * 64;  // wave n-offset in [0,128)

  v8f cAcc[4] = {};

  const int arow = tid >> 2;       // 0..63
  const int acol = (tid & 3) * 4;  // 0,4,8,12
  const float* fA = f + (size_t)(b * Nn + m0 + arow) * Hn + acol;

  for (int k0 = 0; k0 < Hn; k0 += 16) {
    __syncthreads();
    // A: 64x16, coalesced float4, aligned (stride 20 floats = 80B, 16B-mult)
    float4 av = *(const float4*)(fA + k0);
    *(float4*)(&As[arow * 20 + acol]) = av;
    // B: W_lin tile [128 o][16 h] loaded row-major, stored transposed Bs[h][o]
#pragma unroll
    for (int j = 0; j < 2; ++j) {
      int idx = tid + j * 256;
      int br = idx >> 2;        // 0..127
      int bc = (idx & 3) * 4;   // 0,4,8,12
      float4 bv = *(const float4*)(Wlin + (size_t)(o0 + br) * Hn + k0 + bc);
      Bs[(bc + 0) * 136 + br] = bv.x;
      Bs[(bc + 1) * 136 + br] = bv.y;
      Bs[(bc + 2) * 136 + br] = bv.z;
      Bs[(bc + 3) * 136 + br] = bv.w;
    }
    __syncthreads();
#pragma unroll
    for (int ks = 0; ks < 4; ++ks) {
      const int kk = ks * 4 + hi * 2;  // lane's K pair base within the 4-step
      v2f a;
      a.x = As[(wm + lm) * 20 + kk];
      a.y = As[(wm + lm) * 20 + kk + 1];
#pragma unroll
      for (int ct = 0; ct < 4; ++ct) {
        const int n = wn + ct * 16 + lm;
        v2f bb;
        bb.x = Bs[(kk + 0) * 136 + n];
        bb.y = Bs[(kk + 1) * 136 + n];
        cAcc[ct] = __builtin_amdgcn_wmma_f32_16x16x4_f32(
            false, a, false, bb, (short)0, cAcc[ct], false, false);
      }
    }
  }

  // Epilogue: + b_lin, store to workspace. C layout: VGPR r -> M = r + hi*8.
#pragma unroll
  for (int ct = 0; ct < 4; ++ct) {
    const int o = o0 + wn + ct * 16 + lm;
    const float bl = blin[o];
#pragma unroll
    for (int r = 0; r < 8; ++r) {
      const int m = m0 + wm + r + hi * 8;
      wout[(size_t)(b * Nn + m) * NK + o] = cAcc[ct][r] + bl;
    }
  }
}

// ---------------------------------------------------------------------------
// Kernel 2: out = sum_t weight[:,:,t] @ shift_t(k), then LayerNorm over h.
// grid (m_tiles=16, b=64), block 256 (8 waves).
// Workgroup owns 16 output rows x all 1024 h (in registers: 8 v8f per wave).
// K-dim (256 channels) chunked by 16; h split in 2 phases of 512 for LDS.
// ---------------------------------------------------------------------------
__global__ __launch_bounds__(256) void dyconv_ln(
    const float* __restrict__ kin, const float* __restrict__ w,
    const float* __restrict__ gamma, const float* __restrict__ beta,
    float* __restrict__ out) {
  __shared__ float Ks[16 * 520];  // [chan 16][col 514 used]: col = 1 + h_local + (t-1)
  __shared__ float Aw[16 * 52];   // weight chunk [m 16][i_local*3 + t]
  __shared__ float Gs[1024];
  __shared__ float Bt[1024];
  __shared__ float redS[8 * 16];
  __shared__ float redQ[8 * 16];
  __shared__ float muS[16];
  __shared__ float rsS[16];

  const int tid = threadIdx.x;
  const int b = blockIdx.y;
  const int m0 = blockIdx.x * 16;
  const int wave = tid >> 5, lane = tid & 31;
  const int lm = lane & 15, hi2 = lane >> 4;

  // Cache gamma/beta once.
#pragma unroll
  for (int j = 0; j < 4; ++j) {
    int i = tid + j * 256;
    Gs[i] = gamma[i];
    Bt[i] = beta[i];
  }

  v8f acc[8] = {};  // acc[p*4+ct]: h = p*512 + wave*64 + ct*16 + lane_n

  const int aRow = tid >> 4;          // 0..15 (m within tile)
  const int aCol = (tid & 15) * 3;    // 0,3,..,45
  const float* wp = w + (size_t)(b * Nn + m0 + aRow) * NK + aCol;
  const int kRow = tid >> 4;          // 0..15 (channel within chunk)
  const int kCb = (tid & 15) * 32;    // 32 cols per thread
  const int hRow = tid & 15;          // halo row for tid<32

  for (int ic = 0; ic < 16; ++ic) {
    const int i0 = ic * 16;
    __syncthreads();
    // Weight chunk: o = i*3+t is contiguous -> 16x48 floats.
    {
      const float* wq = wp + (size_t)i0 * 3;
      Aw[aRow * 52 + aCol + 0] = wq[0];
      Aw[aRow * 52 + aCol + 1] = wq[1];
      Aw[aRow * 52 + aCol + 2] = wq[2];
    }
#pragma unroll
    for (int p = 0; p < 2; ++p) {
      if (p) __syncthreads();  // compute(p=0) done reading Ks before reload
      // Main slab: cols 1..512  (h_local 0..511)
      {
        const float* kr = kin + (size_t)(b * Nn + i0 + kRow) * Hn + p * 512 + kCb;
#pragma unroll
        for (int j = 0; j < 8; ++j) {
          float4 v = *(const float4*)(kr + j * 4);
          const int cc = kRow * 520 + 1 + kCb + j * 4;  // odd offset: scalar stores
          Ks[cc + 0] = v.x;
          Ks[cc + 1] = v.y;
          Ks[cc + 2] = v.z;
          Ks[cc + 3] = v.w;
        }
      }
      // Halo cols: SAME padding (zeros only at global h=-1 / h=1024).
      if (tid < 32) {
        const float* krh = kin + (size_t)(b * Nn + i0 + hRow) * Hn;
        if (tid < 16) {  // left: h = p*512 - 1
          Ks[hRow * 520 + 0] = (p == 0) ? 0.0f : krh[511];
        } else {         // right: h = p*512 + 512
          Ks[hRow * 520 + 513] = (p == 0) ? krh[512] : 0.0f;
        }
      }
      __syncthreads();
      // 3 taps x 4 k-substeps x 4 h-tiles of WMMA.
#pragma unroll
      for (int t = 0; t < 3; ++t) {
#pragma unroll
        for (int ks = 0; ks < 4; ++ks) {
          const int isub = ks * 4 + hi2 * 2;
          v2f a;
          a.x = Aw[lm * 52 + (isub + 0) * 3 + t];
          a.y = Aw[lm * 52 + (isub + 1) * 3 + t];
#pragma unroll
          for (int ct = 0; ct < 4; ++ct) {
            const int col = (wave * 64 + ct * 16 + lm) + t;  // = 1 + h_local + (t-1)
            v2f bb;
            bb.x = Ks[(isub + 0) * 520 + col];
            bb.y = Ks[(isub + 1) * 520 + col];
            acc[p * 4 + ct] = __builtin_amdgcn_wmma_f32_16x16x4_f32(
                false, a, false, bb, (short)0, acc[p * 4 + ct], false, false);
          }
        }
      }
    }
  }

  // ---- LayerNorm over h (1024) per output row ----
  float s[8], q[8];
#pragma unroll
  for (int r = 0; r < 8; ++r) { s[r] = 0.0f; q[r] = 0.0f; }
#pragma unroll
  for (int u = 0; u < 8; ++u)
#pragma unroll
    for (int r = 0; r < 8; ++r) {
      float v = acc[u][r];
      s[r] += v;
      q[r] += v * v;
    }
  // Reduce across 16 lanes of each half-wave (rows r vs r+8 stay separate).
#pragma unroll
  for (int mask = 1; mask < 16; mask <<= 1)
#pragma unroll
    for (int r = 0; r < 8; ++r) {
      s[r] += __shfl_xor(s[r], mask, 32);
      q[r] += __shfl_xor(q[r], mask, 32);
    }
  if (lm == 0) {  // lanes 0 (rows 0..7) and 16 (rows 8..15)
#pragma unroll
    for (int r = 0; r < 8; ++r) {
      const int row = r + hi2 * 8;
      redS[wave * 16 + row] = s[r];
      redQ[wave * 16 + row] = q[r];
    }
  }
  __syncthreads();
  if (tid < 16) {
    float ss = 0.0f, qq = 0.0f;
#pragma unroll
    for (int wv = 0; wv < 8; ++wv) {
      ss += redS[wv * 16 + tid];
      qq += redQ[wv * 16 + tid];
    }
    const float mu = ss * (1.0f / 1024.0f);
    const float var = qq * (1.0f / 1024.0f) - mu * mu;
    muS[tid] = mu;
    rsS[tid] = rsqrtf(var + LN_EPS);
  }
  __syncthreads();

  // Normalize straight out of the accumulators and store.
#pragma unroll
  for (int u = 0; u < 8; ++u) {
    const int p = u >> 2, ct = u & 3;
    const int h = p * 512 + wave * 64 + ct * 16 + lm;
    const float g = Gs[h], be = Bt[h];
#pragma unroll
    for (int r = 0; r < 8; ++r) {
      const int row = r + hi2 * 8;
      const float v = (acc[u][r] - muS[row]) * rsS[row] * g + be;
      out[(size_t)(b * Nn + m0 + row) * Hn + h] = v;
    }
  }
}

// ---------------------------------------------------------------------------
extern "C" void kernel_launch(void* const* d_in, const int* in_sizes, int n_in,
                              void* d_out, int out_size, void* d_ws, size_t ws_size,
                              hipStream_t stream) {
  const float* f     = (const float*)d_in[0];  // [B,N,H]
  const float* k     = (const float*)d_in[1];  // [B,N,H]
  const float* Wlin  = (const float*)d_in[2];  // [N*K,H]
  const float* blin  = (const float*)d_in[3];  // [N*K]
  const float* gamma = (const float*)d_in[4];  // [H]
  const float* beta  = (const float*)d_in[5];  // [H]
  float* out = (float*)d_out;
  float* wgt = (float*)d_ws;  // needs B*N*NK*4 = 50,331,648 bytes

  (void)in_sizes; (void)n_in; (void)out_size; (void)ws_size;

  // Stage 1: dynamic weight generation GEMM.
  wgen_gemm<<<dim3(NK / 128, Nn / 64, Bn), 256, 0, stream>>>(f, Wlin, blin, wgt);
  // Stage 2: dynamic conv (3 shifted GEMMs) + fused LayerNorm.
  dyconv_ln<<<dim3(Nn / 16, Bn), 256, 0, stream>>>(k, wgt, gamma, beta, out);
}